// BGEM3Model_19739669692451
// MI455X (gfx1250) — compile-verified
//
#include <hip/hip_runtime.h>
#include <stdint.h>

#define B_  32
#define L_  1024
#define H_  1024
#define V_  250002

typedef __attribute__((ext_vector_type(2))) float        v2f;
typedef __attribute__((ext_vector_type(8))) float        v8f;
typedef __attribute__((ext_vector_type(4))) unsigned int v4u;
typedef __attribute__((ext_vector_type(4))) int          v4i;
typedef __attribute__((ext_vector_type(8))) int          v8i;

// ---------------------------------------------------------------------------
// Kernel 1: zero the (B, V) output buffer (32 MB) with wide stores.
// ---------------------------------------------------------------------------
__global__ __launch_bounds__(256) void bgem3_zero_kernel(float4* __restrict__ out4, int n4) {
    int i = blockIdx.x * blockDim.x + threadIdx.x;
    if (i < n4) out4[i] = make_float4(0.f, 0.f, 0.f, 0.f);
}

// ---------------------------------------------------------------------------
// TDM: DMA one 16x128 f32 tile (global, row stride H) into LDS with
// hardware padding: 4 dwords inserted every 128 dwords -> LDS row stride 132.
// D# layout per CDNA5 ISA ch.8 (group0: count/lds/global/type; group1:
// data_size=4B, pad cfg, tensor/tile dims, dim0 stride).
// ---------------------------------------------------------------------------
__device__ __forceinline__ void tdm_load_tile_16x128(uint64_t gaddr, unsigned lds_off) {
    const unsigned TD0 = 1u << 20;   // tensor_dim0: large (never OOB-clip)
    const unsigned TD1 = 1u << 20;   // tensor_dim1: large
    v4u g0;
    g0.x = 1u;                                                    // count=1, user D#
    g0.y = lds_off;                                               // lds_addr (bytes)
    g0.z = (unsigned)gaddr;                                       // global_addr[31:0]
    g0.w = (unsigned)((gaddr >> 32) & 0x01FFFFFFu) | (2u << 30);  // addr[56:32] | type=2
    v8i g1;
    g1[0] = (int)((2u << 16)        // data_size = 4 bytes
                | (1u << 20)        // pad_enable
                | (6u << 22)        // pad_interval: 128 dwords
                | (3u << 25));      // pad_amount: 4 dwords
    g1[1] = (int)((TD0 & 0xFFFFu) << 16);                           // tensor_dim0 lo
    g1[2] = (int)(((TD0 >> 16) & 0xFFFFu) | ((TD1 & 0xFFFFu) << 16));
    g1[3] = (int)(((TD1 >> 16) & 0xFFFFu) | (128u << 16));          // tile_dim0 = 128
    g1[4] = 16;                                                     // tile_dim1=16, tile_dim2=0
    g1[5] = H_;                                                     // tensor_dim0_stride = 1024
    g1[6] = 0;
    g1[7] = 0;
    v4i gz = {0, 0, 0, 0};
#if defined(__clang_major__) && (__clang_major__ >= 23)
    v8i gz8 = {0, 0, 0, 0, 0, 0, 0, 0};
    __builtin_amdgcn_tensor_load_to_lds(g0, g1, gz, gz, gz8, 0);
#else
    __builtin_amdgcn_tensor_load_to_lds(g0, g1, gz, gz, 0);
#endif
}

// ---------------------------------------------------------------------------
// Kernel 2: one wave per 16 tokens. TDM double-buffers 16x128 tiles into LDS;
// dot(hidden, w) via V_WMMA_F32_16X16X4_F32 with w broadcast into all 16 B
// columns (B fetched as a broadcast ds_load_b64 from an LDS copy of w);
// then relu + scatter-max via unsigned atomicMax (weights are >= 0).
// ---------------------------------------------------------------------------
__global__ __launch_bounds__(32) void bgem3_sparse_kernel(
        const float* __restrict__ hidden,   // (B, L, H) f32
        const int*   __restrict__ ids,      // (B, L) i32
        const float* __restrict__ ws,       // (H,) f32
        const float* __restrict__ bs,       // (1,) f32
        float*       __restrict__ out)      // (B, V) f32, pre-zeroed
{
    __shared__ float smem[2 * 2112 + 1024];   // two 16x132 tiles + w copy
    float* wlds = smem + 2 * 2112;

    const int lane     = threadIdx.x;     // 0..31
    const int row      = lane & 15;       // A-matrix row served by this lane
    const int half     = lane >> 4;       // 0: K=0,1   1: K=2,3
    const int tok_base = blockIdx.x * 16; // L%16==0 -> tile stays in one batch
    const int batch    = tok_base >> 10;

    const unsigned lds_base = (unsigned)(uintptr_t)(void*)smem;   // LDS byte offset
    const uint64_t gbase    = (uint64_t)(uintptr_t)hidden
                            + (uint64_t)tok_base * (H_ * 4);

    // kick off DMA of the first tile immediately
    tdm_load_tile_16x128(gbase, lds_base);

    // stage w (1024 f32 = 4 KB) into LDS, coalesced
    const float4* ws4 = (const float4*)ws;
    #pragma unroll
    for (int i = 0; i < 8; ++i)
        ((float4*)wlds)[lane + 32 * i] = ws4[lane + 32 * i];
    __syncthreads();

    const float bias = bs[0];
    v8f c = {};

    for (int i = 0; i < 8; ++i) {
        if (i < 7) {
            // prior reads of the buffer being overwritten must be complete
            asm volatile("s_wait_dscnt 0x0" ::: "memory");
            tdm_load_tile_16x128(gbase + (uint64_t)(i + 1) * 512,
                                 lds_base + (unsigned)(((i + 1) & 1) * 8448));
            // TDM is in-order per wave: cnt<=1 means tile i has landed
            asm volatile("s_wait_tensorcnt 0x1" ::: "memory");
        } else {
            asm volatile("s_wait_tensorcnt 0x0" ::: "memory");
        }

        const float* tile = smem + (i & 1) * 2112;
        const float* wch  = wlds + i * 128;
        #pragma unroll
        for (int kk = 0; kk < 128; kk += 4) {
            // A: lane (row, half) holds A[row][kk+2*half .. +1]; stride 132 dw
            // puts the 32 b64 reads on all 64 banks exactly once.
            v2f a = *(const v2f*)(tile + row * 132 + kk + 2 * half);
            // B: {w[k+2h], w[k+2h+1]} — same K-striping as A, broadcast read.
            v2f b = *(const v2f*)(wch + kk + 2 * half);
            c = __builtin_amdgcn_wmma_f32_16x16x4_f32(
                    /*neg_a=*/false, a, /*neg_b=*/false, b,
                    /*c_mod=*/(short)0, c, /*reuse_a=*/false, /*reuse_b=*/false);
        }
    }

    // D VGPR r: lanes 0-15 -> token tok_base+r, lanes 16-31 -> +8 (all 16
    // columns identical), so lanes 0 and 16 do the per-token scatter.
    if ((lane & 15) == 0) {
        #pragma unroll
        for (int r = 0; r < 8; ++r) {
            float tw = c[r] + bias;            // relu via tw > 0 check
            int tok = tok_base + r + half * 8;
            int id  = ids[tok];
            if (tw > 0.0f && id > 3 && id < V_) {
                // non-negative floats: IEEE order == unsigned bit order
                atomicMax((unsigned int*)(out + (long)batch * V_ + id),
                          __float_as_uint(tw));
            }
        }
    }
}

// ---------------------------------------------------------------------------
extern "C" void kernel_launch(void* const* d_in, const int* in_sizes, int n_in,
                              void* d_out, int out_size, void* d_ws, size_t ws_size,
                              hipStream_t stream) {
    const float* hidden = (const float*)d_in[0];   // (B, L, H) f32
    const int*   ids    = (const int*)  d_in[1];   // (B, L) int
    const float* wsp    = (const float*)d_in[2];   // (H, 1) f32
    const float* bsp    = (const float*)d_in[3];   // (1,) f32
    float*       out    = (float*)d_out;           // (B, V) f32

    // zero-fill output (out_size = 32*250002, divisible by 4)
    int n4 = out_size / 4;
    bgem3_zero_kernel<<<(n4 + 255) / 256, 256, 0, stream>>>((float4*)out, n4);

    // one wave (32 threads) per 16 tokens
    int tiles = (B_ * L_) / 16;   // 2048
    bgem3_sparse_kernel<<<tiles, 32, 0, stream>>>(hidden, ids, wsp, bsp, out);
}